// ISNELayer_67379446940404
// MI455X (gfx1250) — compile-verified
//
#include <hip/hip_runtime.h>
#include <math.h>

// ---------------- problem constants ----------------
#define NN      50000      // nodes
#define EE      1600000    // edges
#define FIN     128
#define FH      128        // hidden = out
#define NH      8          // heads
#define DH      16         // per-head dim
#define ALPHA   0.2f
#define EPSN    1e-12f

#define NTILES  (NN / 16)  // 3125 row tiles (exact)
#define WAVES_PER_BLK 8

typedef float v2f __attribute__((ext_vector_type(2)));
typedef float v8f __attribute__((ext_vector_type(8)));

// fast exp: v_exp_f32 is exp2 on AMD; exp(x) = exp2(x * log2(e)).
// Branchless, no EXEC manipulation, ~2 VALU ops.
__device__ __forceinline__ float fexp(float x) {
    return __builtin_amdgcn_exp2f(x * 1.44269504088896340736f);
}

// ordered-uint encoding of float for atomicMax
__device__ __forceinline__ unsigned fkey(float f) {
    unsigned b = __float_as_uint(f);
    return (b & 0x80000000u) ? ~b : (b | 0x80000000u);
}
__device__ __forceinline__ float funkey(unsigned k) {
    unsigned b = (k & 0x80000000u) ? (k & 0x7FFFFFFFu) : ~k;
    return __uint_as_float(b);
}

// ---------------- init workspace ----------------
__global__ __launch_bounds__(256) void k_init(float* __restrict__ agg,
                                              float* __restrict__ denom,
                                              unsigned* __restrict__ menc) {
    int tid = blockIdx.x * blockDim.x + threadIdx.x;
    if (tid < NN * FH) agg[tid] = 0.0f;
    if (tid < NN * NH) {
        denom[tid] = 0.0f;
        menc[tid]  = 0x007FFFFFu;   // fkey(-inf)
    }
}

// ---------------- fused node GEMMs via f32 WMMA ----------------
// GEMM1: h0 = elu(X @ W1 + b1), tile kept in LDS; GEMM2: h = h0 @ B(Wa)
__global__ __launch_bounds__(256) void k_node_gemms(const float* __restrict__ X,
                                                    const float* __restrict__ W1,
                                                    const float* __restrict__ b1,
                                                    const float* __restrict__ Wa,
                                                    float* __restrict__ h0,
                                                    float* __restrict__ h) {
    __shared__ float smem[WAVES_PER_BLK * 16 * FH];   // 64 KB: 16x128 f32 tile per wave

    const int wave = threadIdx.x >> 5;
    const int lane = threadIdx.x & 31;
    const int tile = blockIdx.x * WAVES_PER_BLK + wave;
    if (tile >= NTILES) return;                       // whole wave exits -> EXEC all-ones for WMMA

    const int half  = lane >> 4;                      // 0 or 1
    const int lcol  = lane & 15;
    const int khalf = half * 2;
    const int row   = tile * 16 + lcol;               // A-fragment row (lane = M)
    float* my = smem + wave * 16 * FH;

    // ---- GEMM1: acc[j] = X_tile(16x128) @ W1(:,16j:16j+16) ----
    v8f acc[8];
    #pragma unroll
    for (int j = 0; j < 8; ++j) acc[j] = v8f{};

    for (int kk = 0; kk < FIN; kk += 4) {
        const int kb = kk + khalf;
        v2f a;
        a.x = X[row * FIN + kb];
        a.y = X[row * FIN + kb + 1];
        #pragma unroll
        for (int j = 0; j < 8; ++j) {
            const int col = j * 16 + lcol;            // B-fragment col (lane = N)
            v2f b;
            b.x = W1[kb * FH + col];
            b.y = W1[(kb + 1) * FH + col];
            acc[j] = __builtin_amdgcn_wmma_f32_16x16x4_f32(
                false, a, false, b, (short)0, acc[j], false, false);
        }
    }

    // bias + ELU (branchless); write tile to LDS (GEMM2 A) and global h0 (residual)
    const int rbase = half * 8;                       // D layout: vgpr r -> M = r + 8*half
    #pragma unroll
    for (int j = 0; j < 8; ++j) {
        const float bias = b1[j * 16 + lcol];
        #pragma unroll
        for (int r = 0; r < 8; ++r) {
            float v = acc[j][r] + bias;
            float en = fexp(v) - 1.0f;                // only used when v<=0 (safe range)
            v = (v > 0.0f) ? v : en;                  // single v_cndmask
            const int m = rbase + r;
            my[m * FH + j * 16 + lcol] = v;
            h0[(tile * 16 + m) * FH + j * 16 + lcol] = v;
        }
    }
    // wave-private LDS region: DS ops complete in-order before DS reads below

    // ---- GEMM2: h_tile = h0_tile @ B where B[f][c] = Wa[c/16][f][c%16] ----
    v8f acc2[8];
    #pragma unroll
    for (int j = 0; j < 8; ++j) acc2[j] = v8f{};

    for (int kk = 0; kk < FH; kk += 4) {
        const int kb = kk + khalf;
        v2f a;
        a.x = my[lcol * FH + kb];                     // A from LDS, lane = M
        a.y = my[lcol * FH + kb + 1];
        #pragma unroll
        for (int j = 0; j < 8; ++j) {
            // col c = j*16 + lcol  ->  Wa[j][kb][lcol] = Wa[j*2048 + kb*16 + lcol]
            v2f b;
            b.x = Wa[j * (FH * DH) + kb * DH + lcol];
            b.y = Wa[j * (FH * DH) + (kb + 1) * DH + lcol];
            acc2[j] = __builtin_amdgcn_wmma_f32_16x16x4_f32(
                false, a, false, b, (short)0, acc2[j], false, false);
        }
    }

    #pragma unroll
    for (int j = 0; j < 8; ++j) {
        #pragma unroll
        for (int r = 0; r < 8; ++r) {
            const int m = rbase + r;
            h[(tile * 16 + m) * FH + j * 16 + lcol] = acc2[j][r];
        }
    }
}

// ---------------- attention logits per node/head ----------------
__global__ __launch_bounds__(256) void k_scores(const float* __restrict__ h,
                                                const float* __restrict__ a_src,
                                                const float* __restrict__ a_dst,
                                                float* __restrict__ ssrc,
                                                float* __restrict__ sdst) {
    int tid = blockIdx.x * blockDim.x + threadIdx.x;
    if (tid >= NN * NH) return;
    const int n = tid >> 3, hd = tid & 7;
    const float* hp = h + (size_t)n * FH + hd * DH;
    const float* as = a_src + hd * DH;
    const float* ad = a_dst + hd * DH;
    float ss = 0.0f, sd = 0.0f;
    #pragma unroll
    for (int d = 0; d < DH; ++d) { float v = hp[d]; ss += v * as[d]; sd += v * ad[d]; }
    ssrc[tid] = ss;
    sdst[tid] = sd;
}

__device__ __forceinline__ float edge_logit(const float* ssrc, const float* sdst,
                                            long long s, long long d, int hd) {
    float ev = ssrc[s * NH + hd] + sdst[d * NH + hd];
    return (ev >= 0.0f) ? ev : ALPHA * ev;
}

// ---------------- segment max ----------------
__global__ __launch_bounds__(256) void k_edge_max(const long long* __restrict__ ei,
                                                  const float* __restrict__ ssrc,
                                                  const float* __restrict__ sdst,
                                                  unsigned* __restrict__ menc) {
    int tid = blockIdx.x * blockDim.x + threadIdx.x;
    if (tid >= EE * NH) return;
    const int e = tid >> 3, hd = tid & 7;
    const long long s = ei[e], d = ei[EE + e];
    const float ev = edge_logit(ssrc, sdst, s, d, hd);
    atomicMax(&menc[d * NH + hd], fkey(ev));
}

// decode max; empty segments (-inf) -> 0
__global__ __launch_bounds__(256) void k_fin_m(unsigned* __restrict__ menc) {
    int tid = blockIdx.x * blockDim.x + threadIdx.x;
    if (tid >= NN * NH) return;
    float f = funkey(menc[tid]);
    if (!isfinite(f)) f = 0.0f;
    ((float*)menc)[tid] = f;
}

// ---------------- segment sum of exp ----------------
__global__ __launch_bounds__(256) void k_edge_sum(const long long* __restrict__ ei,
                                                  const float* __restrict__ ssrc,
                                                  const float* __restrict__ sdst,
                                                  const float* __restrict__ mf,
                                                  float* __restrict__ denom) {
    int tid = blockIdx.x * blockDim.x + threadIdx.x;
    if (tid >= EE * NH) return;
    const int e = tid >> 3, hd = tid & 7;
    const long long s = ei[e], d = ei[EE + e];
    const float ev = edge_logit(ssrc, sdst, s, d, hd);
    const float ex = fexp(ev - mf[d * NH + hd]);      // arg <= 0: fast-path safe
    atomicAdd(&denom[d * NH + hd], ex);
}

// ---------------- weighted scatter-aggregate: 32 lanes per edge ----------------
__global__ __launch_bounds__(256) void k_edge_agg(const long long* __restrict__ ei,
                                                  const float* __restrict__ ssrc,
                                                  const float* __restrict__ sdst,
                                                  const float* __restrict__ mf,
                                                  const float* __restrict__ denom,
                                                  const float* __restrict__ h,
                                                  float* __restrict__ agg) {
    long long tid = (long long)blockIdx.x * blockDim.x + threadIdx.x;
    if (tid >= (long long)EE * 32) return;
    const int e  = (int)(tid >> 5);
    const int t  = (int)(tid & 31);
    const int hd = t >> 2;            // head = feature/16
    const int f0 = t * 4;             // this lane's 4 features
    const long long s = ei[e], d = ei[EE + e];
    const float ev   = edge_logit(ssrc, sdst, s, d, hd);
    const float ex   = fexp(ev - mf[d * NH + hd]);
    const float attn = ex / (denom[d * NH + hd] + 1e-16f);
    const float4 hv = *(const float4*)(h + s * FH + f0);
    float* ap = agg + d * FH + f0;
    atomicAdd(ap + 0, attn * hv.x);
    atomicAdd(ap + 1, attn * hv.y);
    atomicAdd(ap + 2, attn * hv.z);
    atomicAdd(ap + 3, attn * hv.w);
}

// ---------------- residual + L2 normalize: one wave per node ----------------
__global__ __launch_bounds__(256) void k_norm(const float* __restrict__ agg,
                                              const float* __restrict__ h0,
                                              float* __restrict__ out) {
    int gid = blockIdx.x * blockDim.x + threadIdx.x;
    const int node = gid >> 5;
    const int lane = gid & 31;
    if (node >= NN) return;
    const int f0 = lane * 4;
    const float4 a = *(const float4*)(agg + (size_t)node * FH + f0);
    const float4 r = *(const float4*)(h0  + (size_t)node * FH + f0);
    float4 o = make_float4(a.x + r.x, a.y + r.y, a.z + r.z, a.w + r.w);
    float ss = o.x * o.x + o.y * o.y + o.z * o.z + o.w * o.w;
    #pragma unroll
    for (int off = 16; off > 0; off >>= 1) ss += __shfl_xor(ss, off, 32);
    const float scale = 1.0f / fmaxf(sqrtf(ss), EPSN);
    float4 w = make_float4(o.x * scale, o.y * scale, o.z * scale, o.w * scale);
    *(float4*)(out + (size_t)node * FH + f0) = w;
}

// ---------------- launch ----------------
extern "C" void kernel_launch(void* const* d_in, const int* in_sizes, int n_in,
                              void* d_out, int out_size, void* d_ws, size_t ws_size,
                              hipStream_t stream) {
    const float*     x     = (const float*)d_in[0];
    const long long* ei    = (const long long*)d_in[1];   // int64 edge_index [2,E]
    const float*     W1    = (const float*)d_in[2];
    const float*     b1    = (const float*)d_in[3];
    const float*     Wa    = (const float*)d_in[4];
    const float*     a_src = (const float*)d_in[5];
    const float*     a_dst = (const float*)d_in[6];
    float*           out   = (float*)d_out;

    char* w = (char*)d_ws;
    float*    h0    = (float*)w;    w += (size_t)NN * FH * 4;
    float*    h     = (float*)w;    w += (size_t)NN * FH * 4;
    float*    ssrc  = (float*)w;    w += (size_t)NN * NH * 4;
    float*    sdst  = (float*)w;    w += (size_t)NN * NH * 4;
    unsigned* menc  = (unsigned*)w; w += (size_t)NN * NH * 4;   // reused as float m
    float*    denom = (float*)w;    w += (size_t)NN * NH * 4;
    float*    agg   = (float*)w;    w += (size_t)NN * FH * 4;
    float*    mf    = (float*)menc;

    const int B = 256;
    k_init<<<(NN * FH + B - 1) / B, B, 0, stream>>>(agg, denom, menc);

    k_node_gemms<<<(NTILES + WAVES_PER_BLK - 1) / WAVES_PER_BLK, B, 0, stream>>>(
        x, W1, b1, Wa, h0, h);

    k_scores<<<(NN * NH + B - 1) / B, B, 0, stream>>>(h, a_src, a_dst, ssrc, sdst);

    k_edge_max<<<(EE * NH + B - 1) / B, B, 0, stream>>>(ei, ssrc, sdst, menc);

    k_fin_m<<<(NN * NH + B - 1) / B, B, 0, stream>>>(menc);

    k_edge_sum<<<(EE * NH + B - 1) / B, B, 0, stream>>>(ei, ssrc, sdst, mf, denom);

    long long aggThreads = (long long)EE * 32;
    k_edge_agg<<<(unsigned)((aggThreads + B - 1) / B), B, 0, stream>>>(
        ei, ssrc, sdst, mf, denom, h, agg);

    k_norm<<<(NN * 32 + B - 1) / B, B, 0, stream>>>(agg, h0, out);
}